// GNNModule_17575006175795
// MI455X (gfx1250) — compile-verified
//
#include <hip/hip_runtime.h>

// GINEConv x 5 on MI455X (gfx1250, wave32).
//   N=100000 nodes, E=1600000 edges, D=32, L=5.
// Phase 1 (bandwidth-bound): wave-per-edge coalesced gather + relu +
//   global_atomic_add_f32 scatter. Edge indices are wave-uniform -> forced
//   scalar (SMEM) loads via readfirstlane; edge_attr streamed non-temporal
//   so the 12.8MB node state stays L2-resident.
// Phase 2 (matrix): per-wave 16x32 tile MLP with V_WMMA_F32_16X16X4_F32
//   (exact fp32 semantics). Weights stored pair-interleaved in LDS so every
//   B fragment is one conflict-free ds_load_b64.

typedef __attribute__((ext_vector_type(2))) float v2f;
typedef __attribute__((ext_vector_type(8))) float v8f;

#define FD 32      // feature dim
#define WAVES 8    // waves (16-row tiles) per block
#define WSTRIDE 160  // LDS words per 4-k block of pair-interleaved weights

__device__ __forceinline__ int rfl(int v) {
#if defined(__HIP_DEVICE_COMPILE__)
    return __builtin_amdgcn_readfirstlane(v);
#else
    return v;
#endif
}

// ---------------------------------------------------------------- init h
__global__ __launch_bounds__(256) void gine_init_h(
    const float* __restrict__ x, float* __restrict__ h,
    const float* __restrict__ eps, int layer, int n4)
{
    int i = blockIdx.x * blockDim.x + threadIdx.x;
    if (i >= n4) return;
    float s = 1.0f + eps[layer];
    float4 v = ((const float4*)x)[i];
    v.x *= s; v.y *= s; v.z *= s; v.w *= s;
    ((float4*)h)[i] = v;
}

// ---------------------------------------------------------------- scatter
// One wave = one edge: lane d handles feature d. Edge id is wave-uniform ->
// index loads become scalar; feature loads/atomics are 128B coalesced bursts.
__global__ __launch_bounds__(256) void gine_scatter(
    const float* __restrict__ x, const float* __restrict__ ea,
    const long long* __restrict__ src, const long long* __restrict__ dst,
    float* __restrict__ h, int nE)
{
    long long gid = (long long)blockIdx.x * 256 + threadIdx.x;
    int e = rfl((int)(gid >> 5));           // wave-uniform edge id (E < 2^31)
    if (e >= nE) return;
    int d = threadIdx.x & 31;
    long long s = src[e];                   // uniform -> s_load_b64
    long long t = dst[e];
    float v = x[s * FD + d] +
              __builtin_nontemporal_load(&ea[(size_t)e * FD + d]);
    v = v > 0.0f ? v : 0.0f;
    __hip_atomic_fetch_add(&h[t * FD + d], v,
                           __ATOMIC_RELAXED, __HIP_MEMORY_SCOPE_AGENT);
}

// ---------------------------------------------------------------- MLP
// 16x16 f32 WMMA tile: C/D layout = lane -> N (halves -> M+8);
// A 16x4 layout = lane%16 -> M, lanes 16..31 hold K+2 (VGPR pair = K,K+1).
// Weights in LDS are pair-interleaved: block s (k=4s..4s+3) occupies WSTRIDE
// words: [ (k4s,k4s+1) pairs n=0..31 | 32 pad | (k4s+2,k4s+3) pairs n=0..31 ].
// The +96 half-offset puts the two lane halves on disjoint LDS bank sets.
__device__ __forceinline__ int widx(int k, int n) {
    return (k >> 2) * WSTRIDE + ((k >> 1) & 1) * 96 + n * 2 + (k & 1);
}

__device__ __forceinline__ void gemm_tile_32(
    const float (*hs)[34], const float* __restrict__ sW,
    const float* __restrict__ sb,
    int m, int khalf, int ncol, int nhi, v8f& o0, v8f& o1)
{
    float bias0 = sb[ncol];
    float bias1 = sb[16 + ncol];
#pragma unroll
    for (int i = 0; i < 8; ++i) { o0[i] = bias0; o1[i] = bias1; }

#if __has_builtin(__builtin_amdgcn_wmma_f32_16x16x4_f32)
#pragma unroll
    for (int s = 0; s < 8; ++s) {
        int k0 = s * 4 + khalf;                          // even
        int base = s * WSTRIDE + (khalf >> 1) * 96;      // frag block base
        v2f a  = *(const v2f*)&hs[m][k0];                // ds_load_b64
        v2f b0 = *(const v2f*)&sW[base + ncol * 2];      // ds_load_b64
        v2f b1 = *(const v2f*)&sW[base + (16 + ncol) * 2];
        o0 = __builtin_amdgcn_wmma_f32_16x16x4_f32(
                 false, a, false, b0, (short)0, o0, false, false);
        o1 = __builtin_amdgcn_wmma_f32_16x16x4_f32(
                 false, a, false, b1, (short)0, o1, false, false);
    }
#else
    // Scalar fallback producing identical C/D-layout results.
    (void)m; (void)khalf;
#pragma unroll
    for (int i = 0; i < 8; ++i) {
        int row = nhi + i;
        float a0 = o0[i], a1 = o1[i];
        for (int k = 0; k < FD; ++k) {
            float hv = hs[row][k];
            a0 = fmaf(hv, sW[widx(k, ncol)], a0);
            a1 = fmaf(hv, sW[widx(k, 16 + ncol)], a1);
        }
        o0[i] = a0; o1[i] = a1;
    }
#endif
}

__global__ __launch_bounds__(256) void gine_mlp(
    const float* __restrict__ h, const float* __restrict__ W1g,
    const float* __restrict__ b1g, const float* __restrict__ W2g,
    const float* __restrict__ b2g, float* __restrict__ out, int n)
{
    __shared__ __align__(16) float sW1[8 * WSTRIDE];
    __shared__ __align__(16) float sW2[8 * WSTRIDE];
    __shared__ float sb1[FD];
    __shared__ float sb2[FD];
    __shared__ __align__(16) float stage[WAVES][16][34];

    int tid = threadIdx.x;
    for (int i = tid; i < FD * FD; i += 256) {
        int idx = widx(i >> 5, i & 31);     // k = i/32, n = i%32
        sW1[idx] = W1g[i];
        sW2[idx] = W2g[i];
    }
    if (tid < FD) { sb1[tid] = b1g[tid]; sb2[tid] = b2g[tid]; }
    __syncthreads();

    int wave = tid >> 5;
    int lane = tid & 31;
    int ntiles = n >> 4;                     // N % 16 == 0
    int tile = blockIdx.x * WAVES + wave;
    bool valid = tile < ntiles;
    int t = valid ? tile : (ntiles - 1);     // clamp: uniform control flow

    float (*hs)[34] = stage[wave];

    // stage 16x32 tile (lane = column, fully coalesced)
#pragma unroll
    for (int r = 0; r < 16; ++r)
        hs[r][lane] = h[(size_t)(t * 16 + r) * FD + lane];
    __syncthreads();

    int m     = lane & 15;         // A-frag row
    int khalf = (lane >> 4) << 1;  // 0 or 2 (K offset for upper lane half)
    int ncol  = lane & 15;         // B/C/D column
    int nhi   = (lane >> 4) << 3;  // C/D row offset (0 or 8)

    v8f y0, y1;
    gemm_tile_32(hs, sW1, sb1, m, khalf, ncol, nhi, y0, y1);

    __syncthreads();
    // relu + D-layout -> A-layout relayout through LDS
#pragma unroll
    for (int i = 0; i < 8; ++i) {
        hs[nhi + i][ncol]      = fmaxf(y0[i], 0.0f);
        hs[nhi + i][16 + ncol] = fmaxf(y1[i], 0.0f);
    }
    __syncthreads();

    v8f z0, z1;
    gemm_tile_32(hs, sW2, sb2, m, khalf, ncol, nhi, z0, z1);

    if (valid) {
#pragma unroll
        for (int i = 0; i < 8; ++i) {
            size_t row = (size_t)(tile * 16 + nhi + i) * FD;
            out[row + ncol]      = fmaxf(z0[i], 0.0f);  // outer relu
            out[row + 16 + ncol] = fmaxf(z1[i], 0.0f);
        }
    }
}

// ---------------------------------------------------------------- launch
extern "C" void kernel_launch(void* const* d_in, const int* in_sizes, int n_in,
                              void* d_out, int out_size, void* d_ws, size_t ws_size,
                              hipStream_t stream)
{
    const float*     x    = (const float*)d_in[0];
    const float*     ea   = (const float*)d_in[1];
    const long long* ei   = (const long long*)d_in[2];   // int64 [2,E]
    const float*     W1   = (const float*)d_in[3];       // [L,32,32]
    const float*     b1   = (const float*)d_in[4];       // [L,32]
    const float*     W2   = (const float*)d_in[5];
    const float*     b2   = (const float*)d_in[6];
    const float*     eps  = (const float*)d_in[7];       // [L]

    int       N = in_sizes[0] / FD;
    long long E = (long long)in_sizes[1] / FD;
    int       L = in_sizes[7];

    const long long* srcI = ei;
    const long long* dstI = ei + E;

    float* buf0 = (float*)d_ws;
    float* buf1 = buf0 + (size_t)N * FD;
    float* outp = (float*)d_out;

    int n4        = N * (FD / 4);
    int initBlks  = (n4 + 255) / 256;
    long long tot = E * 32;
    int scatBlks  = (int)((tot + 255) / 256);
    int ntiles    = (N + 15) / 16;
    int mlpBlks   = (ntiles + WAVES - 1) / WAVES;

    const float* xcur = x;
    for (int l = 0; l < L; ++l) {
        // last layer computes in place in d_out; earlier layers ping-pong in ws
        float* hbuf = (l == L - 1) ? outp : ((l & 1) ? buf1 : buf0);

        gine_init_h<<<initBlks, 256, 0, stream>>>(xcur, hbuf, eps, l, n4);
        gine_scatter<<<scatBlks, 256, 0, stream>>>(xcur, ea, srcI, dstI, hbuf, (int)E);
        gine_mlp<<<mlpBlks, 256, 0, stream>>>(
            hbuf, W1 + (size_t)l * FD * FD, b1 + (size_t)l * FD,
            W2 + (size_t)l * FD * FD, b2 + (size_t)l * FD, hbuf, N);

        xcur = hbuf;
    }
}